// Integrator_54460185313483
// MI455X (gfx1250) — compile-verified
//
#include <hip/hip_runtime.h>
#include <hip/hip_bf16.h>
#include <stdint.h>

// Problem constants (from reference setup_inputs)
#define DIM_XS 256
#define DIM_YS 256
#define DIM_ZS 256
static constexpr int NVOX = DIM_XS * DIM_YS * DIM_ZS;   // 16,777,216
static constexpr int NPTS = 480 * 640 * 9;              // 2,764,800
static constexpr int NENT = NPTS * 8;                   // 22,118,400

typedef float              f32x4 __attribute__((ext_vector_type(4)));
typedef int                i32x4 __attribute__((ext_vector_type(4)));
typedef int                i32x8 __attribute__((ext_vector_type(8)));
typedef unsigned int       u32x4 __attribute__((ext_vector_type(4)));
typedef unsigned long long u64x2 __attribute__((ext_vector_type(2)));

// ---------------------------------------------------------------------------
// Kernel 1: zero the scratch accumulators.
// d_ws layout: Wacc[NVOX] f32 | Uacc[NVOX] f32 | Pmax[NVOX] u64 = 16 B/voxel.
// Temporal stores: these lines are about to be hammered by atomics.
// ---------------------------------------------------------------------------
__global__ void k_init_ws(u32x4* __restrict__ ws16) {
    int i = blockIdx.x * blockDim.x + threadIdx.x;
    if (i < NVOX) {
        u32x4 z = (u32x4)(0u);
        ws16[i] = z;
    }
}

// ---------------------------------------------------------------------------
// Kernel 2: scatter. One thread per (point, neighbor) entry.
// NT loads for the 353 MB single-use index/weight streams so the 268 MB
// accumulator set keeps L2 residency for the atomics.
// ---------------------------------------------------------------------------
__global__ void k_scatter(const float* __restrict__ values,
                          const int*   __restrict__ indices,
                          const float* __restrict__ weights,
                          const int*   __restrict__ semantics,
                          const float* __restrict__ scores,
                          float* __restrict__ Wacc,
                          float* __restrict__ Uacc,
                          unsigned long long* __restrict__ Pmax) {
    int e = blockIdx.x * blockDim.x + threadIdx.x;
    if (e >= NENT) return;

    const int* ip = indices + (size_t)e * 3;
    int ix = __builtin_nontemporal_load(ip + 0);
    int iy = __builtin_nontemporal_load(ip + 1);
    int iz = __builtin_nontemporal_load(ip + 2);
    float w = __builtin_nontemporal_load(weights + e);

    bool valid = ((unsigned)ix < (unsigned)DIM_XS) &
                 ((unsigned)iy < (unsigned)DIM_YS) &
                 ((unsigned)iz < (unsigned)DIM_ZS);
    if (!valid) return;

    int flat = (ix * DIM_YS + iy) * DIM_ZS + iz;
    int p = e >> 3;

    float v  = values[p];
    float sc = scores[p];
    int  sem = semantics[p];

    atomicAdd(Wacc + flat, w);          // segment_sum(w)
    atomicAdd(Uacc + flat, w * v);      // segment_sum(w*v)

    // argmax-by-score label fusion: scores are positive floats, so float-bit
    // order == integer order. Nonzero pk doubles as the "touched" flag.
    unsigned long long pk =
        ((unsigned long long)__float_as_uint(sc) << 32) | (unsigned)sem;
    atomicMax(Pmax + flat, pk);
}

// ---------------------------------------------------------------------------
// TDM helper: issue a 1-D tile DMA (global -> LDS) via the Tensor Data Mover.
// Descriptor packing per CDNA5 ISA 8.3/8.4:
//   group0: [1:0]=count=1, [63:32]=lds_addr, [120:64]=global_addr, [127:126]=2
//   group1: [17:16]=data_size, [79:48]=tensor_dim0, [111:80]=tensor_dim1=1,
//           [127:112]=tile_dim0, [207:160]=tensor_dim0_stride
// groups 2/3 zero (tensor <= 2D). Completion tracked with TENSORcnt.
// This toolchain exposes the 6-arg builtin:
//   (u32x4 g0, i32x8 g1, i32x4 g2, i32x4 g3, i32x8 extra, i32 cpol)
// ---------------------------------------------------------------------------
__device__ __forceinline__ uint32_t lds_byte_off(const void* p) {
    // LDS aperture: generic addr[31:0] is the LDS byte offset.
    return (uint32_t)(uintptr_t)p;
}

__device__ __forceinline__ void tdm_load_1d(const void* gsrc, uint32_t lds_off,
                                            uint32_t nelem, uint32_t dsz_code) {
    uint64_t ga = (uint64_t)(uintptr_t)gsrc;

    u32x4 g0;
    g0.x = 1u;                                               // count=1
    g0.y = lds_off;                                          // lds_addr
    g0.z = (uint32_t)ga;                                     // global_addr lo
    g0.w = (uint32_t)((ga >> 32) & 0x01FFFFFFu) | (2u << 30);// addr hi | type=2

    i32x8 g1;
    g1[0] = (int)(dsz_code << 16);                 // data_size (0/1/2/3)
    g1[1] = (int)((nelem & 0xFFFFu) << 16);        // tensor_dim0[15:0]
    g1[2] = (int)(((nelem >> 16) & 0xFFFFu)        // tensor_dim0[31:16]
                  | (1u << 16));                   // tensor_dim1 = 1
    g1[3] = (int)((nelem & 0xFFFFu) << 16);        // tile_dim0 = nelem
    g1[4] = 0;                                     // tile_dim1/2 unused
    g1[5] = (int)nelem;                            // tensor_dim0_stride lo
    g1[6] = 0;
    g1[7] = 0;

    i32x4 z4 = (i32x4)(0);
    i32x8 z8 = (i32x8)(0);
    __builtin_amdgcn_tensor_load_to_lds(g0, g1, z4, z4, z8, 0);
}

// ---------------------------------------------------------------------------
// Kernel 3: finalize, TDM-staged. Each 256-thread block owns a 1024-voxel
// tile (32 KB LDS). Wave 0 DMAs all 7 input streams to LDS via the Tensor
// Data Mover, waits on TENSORcnt, barriers; then every thread computes 4
// voxels from LDS and streams the 4 outputs with NT 128-bit stores.
// ---------------------------------------------------------------------------
#define TILE 1024

__global__ void k_finalize(const float* __restrict__ vvol,
                           const float* __restrict__ wvol,
                           const int*   __restrict__ svol,
                           const float* __restrict__ scvol,
                           const float* __restrict__ Wacc,
                           const float* __restrict__ Uacc,
                           const unsigned long long* __restrict__ Pmax,
                           float* __restrict__ out_v,
                           float* __restrict__ out_w,
                           int*   __restrict__ out_sem,
                           float* __restrict__ out_sc) {
    __shared__ float              s_v[TILE];
    __shared__ float              s_w[TILE];
    __shared__ int                s_s[TILE];
    __shared__ float              s_sc[TILE];
    __shared__ float              s_W[TILE];
    __shared__ float              s_U[TILE];
    __shared__ unsigned long long s_P[TILE];

    const int vb = blockIdx.x * TILE;   // tile base voxel

    if (threadIdx.x < 32) {             // wave 0 issues the tile DMAs
        tdm_load_1d(vvol  + vb, lds_byte_off(s_v),  TILE, 2u);
        tdm_load_1d(wvol  + vb, lds_byte_off(s_w),  TILE, 2u);
        tdm_load_1d(svol  + vb, lds_byte_off(s_s),  TILE, 2u);
        tdm_load_1d(scvol + vb, lds_byte_off(s_sc), TILE, 2u);
        tdm_load_1d(Wacc  + vb, lds_byte_off(s_W),  TILE, 2u);
        tdm_load_1d(Uacc  + vb, lds_byte_off(s_U),  TILE, 2u);
        tdm_load_1d(Pmax  + vb, lds_byte_off(s_P),  TILE, 3u);
        __builtin_amdgcn_s_wait_tensorcnt(0);
    }
    __syncthreads();

    const int j0 = threadIdx.x * 4;     // 4 voxels per thread

    f32x4 vo  = *(const f32x4*)&s_v[j0];
    f32x4 wo  = *(const f32x4*)&s_w[j0];
    i32x4 so  = *(const i32x4*)&s_s[j0];
    f32x4 sco = *(const f32x4*)&s_sc[j0];
    f32x4 Wv  = *(const f32x4*)&s_W[j0];
    f32x4 Uv  = *(const f32x4*)&s_U[j0];
    u64x2 p01 = *(const u64x2*)&s_P[j0];
    u64x2 p23 = *(const u64x2*)&s_P[j0 + 2];

    f32x4 rv, rw, rsc;
    i32x4 rsem;

#pragma unroll
    for (int j = 0; j < 4; ++j) {
        unsigned long long pk = (j < 2) ? p01[j] : p23[j - 2];
        bool touched = (pk != 0ull);

        float w_old = wo[j];
        float v_old = vo[j];
        float wu    = w_old + Wv[j];
        float denom = (wu == 0.0f) ? 1.0f : wu;
        float v_new = __builtin_fmaf(w_old, v_old, Uv[j]) / denom;

        rv[j] = touched ? v_new : v_old;
        rw[j] = touched ? wu    : w_old;

        float sc_c = __uint_as_float((unsigned)(pk >> 32));
        int   lab  = (int)(unsigned)(pk & 0xffffffffull);
        bool  take = touched && (sc_c > sco[j]);
        rsc[j]  = take ? sc_c : sco[j];
        rsem[j] = take ? lab  : so[j];
    }

    const int ovec = vb / 4 + threadIdx.x;   // f32x4 index into outputs
    __builtin_nontemporal_store(rv,   (f32x4*)out_v   + ovec);
    __builtin_nontemporal_store(rw,   (f32x4*)out_w   + ovec);
    __builtin_nontemporal_store(rsem, (i32x4*)out_sem + ovec);
    __builtin_nontemporal_store(rsc,  (f32x4*)out_sc  + ovec);
}

// ---------------------------------------------------------------------------
extern "C" void kernel_launch(void* const* d_in, const int* in_sizes, int n_in,
                              void* d_out, int out_size, void* d_ws, size_t ws_size,
                              hipStream_t stream) {
    const float* values    = (const float*)d_in[0];
    const int*   indices   = (const int*)  d_in[1];
    const float* weights   = (const float*)d_in[2];
    const int*   semantics = (const int*)  d_in[3];
    const float* scores    = (const float*)d_in[4];
    const float* vvol      = (const float*)d_in[5];
    const float* wvol      = (const float*)d_in[6];
    const int*   svol      = (const int*)  d_in[7];
    const float* scvol     = (const float*)d_in[8];

    // scratch layout: Wacc | Uacc | Pmax  (16 B per voxel = 268 MB)
    float* Wacc = (float*)d_ws;
    float* Uacc = Wacc + NVOX;
    unsigned long long* Pmax =
        (unsigned long long*)((char*)d_ws + (size_t)NVOX * 8);

    // output layout: new_v | new_w | new_sem | new_sc (reference return order)
    float* out_v   = (float*)d_out;
    float* out_w   = out_v + NVOX;
    int*   out_sem = (int*)(out_v + 2 * (size_t)NVOX);
    float* out_sc  = out_v + 3 * (size_t)NVOX;

    const int B = 256;  // 8 wave32s per block

    k_init_ws<<<(NVOX + B - 1) / B, B, 0, stream>>>((u32x4*)d_ws);

    k_scatter<<<(NENT + B - 1) / B, B, 0, stream>>>(
        values, indices, weights, semantics, scores, Wacc, Uacc, Pmax);

    k_finalize<<<NVOX / TILE, B, 0, stream>>>(
        vvol, wvol, svol, scvol, Wacc, Uacc, Pmax,
        out_v, out_w, out_sem, out_sc);
}